// ModelDEP_11433202942193
// MI455X (gfx1250) — compile-verified
//
#include <hip/hip_runtime.h>

// Problem constants
#define BB 16
#define LL 128
#define DIN 512
#define HH 256
#define NTAGS 45
#define ROWS (BB * LL)   // 2048

typedef float v2f __attribute__((ext_vector_type(2)));
typedef float v8f __attribute__((ext_vector_type(8)));

// ---------------------------------------------------------------------------
// fp32 WMMA GEMM with compile-time K/N: C[M,N] = act(A[M,K] @ B[K,N] + bias).
// One wave computes one 16x16 tile via V_WMMA_F32_16X16X4_F32, K-loop step 4.
// A-fragment (16x4 f32): lanes 0-15 m=lane, VGPR0=K0 VGPR1=K1;
//                        lanes 16-31 m=lane-16, VGPR0=K2 VGPR1=K3.
// B-fragment (4x16 f32): mirrored (VGPR0 = row k0+2*hi, VGPR1 = row k0+2*hi+1).
// C/D (16x16 f32): c[i] -> row i + 8*(lane/16), col lane%16.
// Compile-time strides let the compiler fold B's row stride (N*4 bytes) into
// immediate global_load offsets / induction-variable adds instead of v_mul_u64.
// ---------------------------------------------------------------------------
template <int K, int N, int RELU, int BIAS>
__global__ __launch_bounds__(256)
void wmma_gemm_f32(const float* __restrict__ A, const float* __restrict__ B,
                   const float* __restrict__ bias, float* __restrict__ C)
{
    const int wave = threadIdx.x >> 5;          // 0..7
    const int lane = threadIdx.x & 31;
    constexpr int tilesN = N >> 4;
    const int tile = blockIdx.x * 8 + wave;
    const int tm = tile / tilesN;
    const int tn = tile % tilesN;

    const int m  = lane & 15;
    const int hi = lane >> 4;                   // 0 or 1

    // Per-lane base pointers; loop indices become constant-stride offsets.
    const float* __restrict__ Aptr = A + (size_t)(tm * 16 + m) * K + 2 * hi;
    const float* __restrict__ Bptr = B + (size_t)(2 * hi) * N + tn * 16 + m;

    v8f c = {0.f, 0.f, 0.f, 0.f, 0.f, 0.f, 0.f, 0.f};

#pragma unroll 4
    for (int k0 = 0; k0 < K; k0 += 4) {
        v2f a, b;
        a.x = Aptr[k0];                          // contiguous pair -> b64 load
        a.y = Aptr[k0 + 1];
        b.x = Bptr[(size_t)k0 * N];              // stride-reduced to +4*N/iter
        b.y = Bptr[(size_t)k0 * N + N];
        c = __builtin_amdgcn_wmma_f32_16x16x4_f32(
                /*neg_a=*/false, a, /*neg_b=*/false, b,
                /*c_mod=*/(short)0, c, /*reuse_a=*/false, /*reuse_b=*/false);
    }

    const int col = tn * 16 + m;
    const float bval = BIAS ? bias[col] : 0.f;
    float* __restrict__ Cptr = C + (size_t)(tm * 16 + 8 * hi) * N + col;
#pragma unroll
    for (int i = 0; i < 8; ++i) {
        float v = c[i] + bval;
        if (RELU) v = fmaxf(v, 0.f);
        Cptr[(size_t)i * N] = v;
    }
}

// ---------------------------------------------------------------------------
// hroot[n] = sum_k root[k] * Wb[k,n]   (Wb = Wp + H*H), 256 threads.
// ---------------------------------------------------------------------------
__global__ __launch_bounds__(256)
void root_proj(const float* __restrict__ root, const float* __restrict__ Wb,
               float* __restrict__ hroot)
{
    const int n = threadIdx.x;
    float acc = 0.f;
#pragma unroll 8
    for (int k = 0; k < HH; ++k) acc += root[k] * Wb[(size_t)k * HH + n];
    hroot[n] = acc;
}

// ---------------------------------------------------------------------------
// Fused biaffine scoring + arc log-softmax + label head + label log-softmax.
// One workgroup per (b,i) row; never materializes pairs_repr (270 MB saved).
// contrib[row] = mask * (arc_ce + lab_ce)
// ---------------------------------------------------------------------------
__global__ __launch_bounds__(256)
void arc_label_loss(const float* __restrict__ ha, const float* __restrict__ hb,
                    const float* __restrict__ hroot, const float* __restrict__ bp,
                    const float* __restrict__ W_arc, const float* __restrict__ b_arc,
                    const float* __restrict__ W_lab, const float* __restrict__ b_lab,
                    const int* __restrict__ slen, const int* __restrict__ arcs,
                    const int* __restrict__ labels, float* __restrict__ contrib)
{
    __shared__ float av[HH];         // ha[b,i,:] + bp
    __shared__ float hr[HH];         // hroot
    __shared__ float wa[HH];         // W_arc
    __shared__ float sel[HH];        // selected pair vector
    __shared__ float scores[LL + 1]; // 129 arc logits
    __shared__ float lgts[NTAGS];
    __shared__ float sh_arc_ce, sh_lab_ce;

    const int row = blockIdx.x;      // 0..2047
    const int b = row >> 7;
    const int i = row & 127;
    const int t = threadIdx.x;
    const int wave = t >> 5;
    const int lane = t & 31;

    av[t] = ha[(size_t)row * HH + t] + bp[t];
    hr[t] = hroot[t];
    wa[t] = W_arc[t];
    __syncthreads();

    // arc scores: each wave handles j = wave, wave+8, ...
    for (int j = wave; j < LL + 1; j += 8) {
        const float* __restrict__ hbj =
            (j == 0) ? hr : &hb[(size_t)(b * LL + (j - 1)) * HH];
        float s = 0.f;
#pragma unroll
        for (int k = lane; k < HH; k += 32) {
            float v = av[k] + hbj[k];
            v = fmaxf(v, 0.f);
            s += v * wa[k];
        }
        for (int off = 16; off; off >>= 1) s += __shfl_xor(s, off, 32);
        if (lane == 0) scores[j] = s + b_arc[0];
    }
    __syncthreads();

    // arc log-softmax (wave 0)
    if (wave == 0) {
        float mx = -3.4e38f;
        for (int j = lane; j < LL + 1; j += 32) mx = fmaxf(mx, scores[j]);
        for (int off = 16; off; off >>= 1) mx = fmaxf(mx, __shfl_xor(mx, off, 32));
        float se = 0.f;
        for (int j = lane; j < LL + 1; j += 32) se += __expf(scores[j] - mx);
        for (int off = 16; off; off >>= 1) se += __shfl_xor(se, off, 32);
        if (lane == 0) {
            const int arc = arcs[row];
            sh_arc_ce = -(scores[arc] - mx - __logf(se));
        }
    }

    // label head: sel = relu(av + hb_row[arc])
    const int arc = arcs[row];
    const float* __restrict__ hba =
        (arc == 0) ? hr : &hb[(size_t)(b * LL + (arc - 1)) * HH];
    sel[t] = fmaxf(av[t] + hba[t], 0.f);
    __syncthreads();

    if (t < NTAGS) {
        float s = b_lab[t];
#pragma unroll 8
        for (int k = 0; k < HH; ++k) s += sel[k] * W_lab[(size_t)k * NTAGS + t];
        lgts[t] = s;
    }
    __syncthreads();

    // label log-softmax (wave 0)
    if (wave == 0) {
        float mx = -3.4e38f;
        for (int j = lane; j < NTAGS; j += 32) mx = fmaxf(mx, lgts[j]);
        for (int off = 16; off; off >>= 1) mx = fmaxf(mx, __shfl_xor(mx, off, 32));
        float se = 0.f;
        for (int j = lane; j < NTAGS; j += 32) se += __expf(lgts[j] - mx);
        for (int off = 16; off; off >>= 1) se += __shfl_xor(se, off, 32);
        if (lane == 0) {
            const int lb = labels[row];
            sh_lab_ce = -(lgts[lb] - mx - __logf(se));
        }
    }
    __syncthreads();

    if (t == 0) {
        const float mask = (i < slen[b]) ? 1.f : 0.f;
        contrib[row] = mask * (sh_arc_ce + sh_lab_ce);
    }
}

// ---------------------------------------------------------------------------
// Deterministic reduction of 2048 contributions + denom, final scale.
// ---------------------------------------------------------------------------
__global__ __launch_bounds__(256)
void finalize_loss(const float* __restrict__ contrib, const int* __restrict__ slen,
                   float* __restrict__ out)
{
    __shared__ float sh[256];
    const int t = threadIdx.x;
    float s = 0.f;
    for (int r = t; r < ROWS; r += 256) s += contrib[r];
    sh[t] = s;
    __syncthreads();
    for (int off = 128; off; off >>= 1) {
        if (t < off) sh[t] += sh[t + off];
        __syncthreads();
    }
    if (t == 0) {
        int d = 0;
        for (int b = 0; b < BB; ++b) {
            int v = slen[b];
            if (v > LL) v = LL;
            d += v;
        }
        float denom = (float)d;
        if (denom < 1.f) denom = 1.f;
        out[0] = sh[0] * 0.5f / denom;
    }
}

// ---------------------------------------------------------------------------
extern "C" void kernel_launch(void* const* d_in, const int* in_sizes, int n_in,
                              void* d_out, int out_size, void* d_ws, size_t ws_size,
                              hipStream_t stream)
{
    const float* X      = (const float*)d_in[0];   // (16,128,512)
    const int*   slen   = (const int*)  d_in[1];   // (16,)
    const int*   arcs   = (const int*)  d_in[2];   // (16,128)
    const int*   labels = (const int*)  d_in[3];   // (16,128)
    const float* W1     = (const float*)d_in[4];   // (512,256)
    const float* b1     = (const float*)d_in[5];   // (256,)
    const float* root   = (const float*)d_in[6];   // (256,)
    const float* Wp     = (const float*)d_in[7];   // (512,256)
    const float* bp     = (const float*)d_in[8];   // (256,)
    const float* W_arc  = (const float*)d_in[9];   // (256,1)
    const float* b_arc  = (const float*)d_in[10];  // (1,)
    const float* W_lab  = (const float*)d_in[11];  // (256,45)
    const float* b_lab  = (const float*)d_in[12];  // (45,)
    float* out = (float*)d_out;

    // workspace layout (floats)
    float* ws      = (float*)d_ws;
    float* hidden  = ws;                         // 2048*256
    float* ha      = hidden + ROWS * HH;         // 2048*256
    float* hb      = ha     + ROWS * HH;         // 2048*256
    float* hroot   = hb     + ROWS * HH;         // 256
    float* contrib = hroot  + HH;                // 2048

    constexpr int gemm_blocks = (ROWS / 16) * (HH / 16) / 8;   // 256

    // 1) hidden = relu(X @ W1 + b1)        M=2048 N=256 K=512
    wmma_gemm_f32<DIN, HH, 1, 1><<<gemm_blocks, 256, 0, stream>>>(X, W1, b1, hidden);

    // 2) ha = hidden @ Wa ; hb = hidden @ Wb    (Wa = Wp[:256], Wb = Wp[256:])
    wmma_gemm_f32<HH, HH, 0, 0><<<gemm_blocks, 256, 0, stream>>>(hidden, Wp, nullptr, ha);
    wmma_gemm_f32<HH, HH, 0, 0><<<gemm_blocks, 256, 0, stream>>>(hidden, Wp + HH * HH, nullptr, hb);

    // 3) hroot = root @ Wb
    root_proj<<<1, 256, 0, stream>>>(root, Wp + HH * HH, hroot);

    // 4) fused biaffine scores + CE per row
    arc_label_loss<<<ROWS, 256, 0, stream>>>(
        ha, hb, hroot, bp, W_arc, b_arc, W_lab, b_lab,
        slen, arcs, labels, contrib);

    // 5) deterministic reduction -> scalar loss
    finalize_loss<<<1, 256, 0, stream>>>(contrib, slen, out);
}